// GCNlayer_86388972192305
// MI455X (gfx1250) — compile-verified
//
#include <hip/hip_runtime.h>
#include <math.h>

#define NNODES 8192
#define INDIM  512
#define HID    256
#define KB     32

typedef float v2f __attribute__((ext_vector_type(2)));
typedef float v8f __attribute__((ext_vector_type(8)));

__device__ __forceinline__ v8f wmma_f32(v2f a, v2f b, v8f c) {
  // V_WMMA_F32_16X16X4_F32 : D = A(16x4) * B(4x16) + C(16x16), all fp32
  return __builtin_amdgcn_wmma_f32_16x16x4_f32(false, a, false, b, (short)0, c,
                                               false, false);
}

// CDNA5 async memory->LDS copy, 16B per lane, tracked by ASYNCcnt.
__device__ __forceinline__ void async_copy_b128(void* lds_dst, const void* gsrc) {
  unsigned lds_off = (unsigned)(uintptr_t)lds_dst;  // low 32 bits = LDS offset
  unsigned long long ga = (unsigned long long)(uintptr_t)gsrc;
  asm volatile("global_load_async_to_lds_b128 %0, %1, off"
               :
               : "v"(lds_off), "v"(ga)
               : "memory");
}
__device__ __forceinline__ void wait_async_le9() {
  asm volatile("s_wait_asynccnt 9" ::: "memory");
}
__device__ __forceinline__ void wait_async_le0() {
  asm volatile("s_wait_asynccnt 0" ::: "memory");
}

// ---------------------------------------------------------------------------
// Kernel 1: dinv[row] = (sum(adj[row,:]) + 1)^(-1/2)
// ---------------------------------------------------------------------------
__global__ __launch_bounds__(256) void rowsum_dinv_kernel(
    const float* __restrict__ adj, float* __restrict__ dinv) {
  __shared__ float red[256];
  const int row = blockIdx.x;
  const int tid = threadIdx.x;
  const float4* rp = reinterpret_cast<const float4*>(adj + (size_t)row * NNODES);
  float s = 0.f;
#pragma unroll
  for (int i = 0; i < NNODES / (256 * 4); ++i) {
    float4 v = rp[i * 256 + tid];
    s += (v.x + v.y) + (v.z + v.w);
  }
  red[tid] = s;
  __syncthreads();
  for (int off = 128; off > 0; off >>= 1) {
    if (tid < off) red[tid] += red[tid + off];
    __syncthreads();
  }
  if (tid == 0) {
    float deg = red[0] + 1.0f;
    dinv[row] = deg > 0.f ? rsqrtf(deg) : 0.f;
  }
}

// ---------------------------------------------------------------------------
// Kernel 2: out[M,256] = dinv[:,None] * (A[M,K] @ W[K,256] + bias)
// Block: 256 threads = 8 waves; block tile 32 rows x 256 cols.
// Wave (strip = w/4, colgrp = w%4) computes 16 x 64 via 4 WMMA C tiles.
// ---------------------------------------------------------------------------
__global__ __launch_bounds__(256) void dense_gemm_kernel(
    const float* __restrict__ A, const float* __restrict__ W,
    const float* __restrict__ bias, const float* __restrict__ dinv,
    float* __restrict__ out, int K) {
  const int tid = threadIdx.x;
  const int wave = tid >> 5, lane = tid & 31;
  const int strip = wave >> 2, cg = wave & 3;
  const int l = lane & 15;             // row (A) / col (B) within tile
  const int kofs = (lane >> 4) << 1;   // lanes 16-31 hold K+2,K+3
  const int row0 = blockIdx.x * 32 + strip * 16;
  const int col0 = cg * 64;

  const v8f vzero = {0.f, 0.f, 0.f, 0.f, 0.f, 0.f, 0.f, 0.f};
  v8f acc[4];
#pragma unroll
  for (int t = 0; t < 4; ++t) acc[t] = vzero;

  const float* Arow = A + (size_t)(row0 + l) * K;
  for (int k = 0; k < K; k += 4) {
    v2f af;
    af.x = Arow[k + kofs];
    af.y = Arow[k + kofs + 1];
#pragma unroll
    for (int t = 0; t < 4; ++t) {
      const int col = col0 + t * 16 + l;
      v2f bf;
      bf.x = W[(size_t)(k + kofs) * HID + col];
      bf.y = W[(size_t)(k + kofs + 1) * HID + col];
      acc[t] = wmma_f32(af, bf, acc[t]);
    }
  }

  const int rsel = (lane >> 4) << 3;  // C layout: lanes 16-31 -> M += 8
#pragma unroll
  for (int t = 0; t < 4; ++t) {
    const int col = col0 + t * 16 + l;
#pragma unroll
    for (int v = 0; v < 8; ++v) {
      const int row = row0 + v + rsel;
      out[(size_t)row * HID + col] = dinv[row] * (acc[t][v] + bias[col]);
    }
  }
}

// ---------------------------------------------------------------------------
// Kernel 3: t = dinv[:,None] * (adj @ hs + hs)
//   fuse==1 : out = LayerNorm(gelu_exact(t)) * gamma + beta
//   fuse==0 : out = t
// Block tile: 32 rows x 256 cols; K staged via double-buffered async->LDS
// copies (GLOBAL_LOAD_ASYNC_TO_LDS_B128 / ASYNCcnt) in steps of KB=32.
// ---------------------------------------------------------------------------
__global__ __launch_bounds__(256) void adj_gemm_kernel(
    const float* __restrict__ adj, const float* __restrict__ hs,
    const float* __restrict__ dinv, const float* __restrict__ gamma,
    const float* __restrict__ beta, float* __restrict__ out, int fuse) {
  // 36-float row stride: 144B (16B aligned for B128 async) and conflict-free.
  __shared__ float lA[2][32][36];     // adj tiles (double buffered)
  __shared__ float lB[2][KB][HID];    // hs tiles; buf0 reused as gelu tile
  __shared__ float redS[32][8], redS2[32][8];
  __shared__ float mu_s[32], rstd_s[32];

  const int tid = threadIdx.x;
  const int wave = tid >> 5, lane = tid & 31;
  const int strip = wave >> 2, cg = wave & 3;
  const int l = lane & 15;
  const int kofs = (lane >> 4) << 1;
  const int m0 = blockIdx.x * 32;
  const int col0 = cg * 64;

  const v8f vzero = {0.f, 0.f, 0.f, 0.f, 0.f, 0.f, 0.f, 0.f};
  v8f acc[4];
#pragma unroll
  for (int t = 0; t < 4; ++t) acc[t] = vzero;

  const int la_r = tid >> 3;            // adj tile row this thread stages
  const int la_c = (tid & 7) << 2;      // 4 floats (16B) per thread per row

  // Issue one tile's worth of async copies (9 ops per thread, in-order done).
  auto issue_tile = [&](int buf, int k0) {
    async_copy_b128(&lA[buf][la_r][la_c],
                    adj + (size_t)(m0 + la_r) * NNODES + k0 + la_c);
#pragma unroll
    for (int i = 0; i < 8; ++i) {
      const int c = i * 256 + tid;      // float4 chunk index, 64 per row
      const int r = c >> 6;
      const int cc = (c & 63) << 2;
      async_copy_b128(&lB[buf][r][cc], hs + (size_t)(k0 + r) * HID + cc);
    }
  };

  issue_tile(0, 0);
  int cur = 0;
  for (int k0 = 0; k0 < NNODES; k0 += KB) {
    if (k0 + KB < NNODES) {
      issue_tile(cur ^ 1, k0 + KB);   // prefetch next tile (9 ops in flight)
      wait_async_le9();               // current tile's 9 ops have completed
    } else {
      wait_async_le0();
    }
    __syncthreads();                  // all waves' current tile landed
#pragma unroll
    for (int kk = 0; kk < KB; kk += 4) {
      v2f af;
      af.x = lA[cur][strip * 16 + l][kk + kofs];
      af.y = lA[cur][strip * 16 + l][kk + kofs + 1];
#pragma unroll
      for (int t = 0; t < 4; ++t) {
        const int col = col0 + t * 16 + l;
        v2f bf;
        bf.x = lB[cur][kk + kofs][col];
        bf.y = lB[cur][kk + kofs + 1][col];
        acc[t] = wmma_f32(af, bf, acc[t]);
      }
    }
    __syncthreads();                  // buffer free for iteration k0+2*KB
    cur ^= 1;
  }

  const int rsel = (lane >> 4) << 3;
  if (!fuse) {
#pragma unroll
    for (int t = 0; t < 4; ++t) {
      const int col = col0 + t * 16 + l;
#pragma unroll
      for (int v = 0; v < 8; ++v) {
        const int rl = strip * 16 + v + rsel;
        const int grow = m0 + rl;
        float val = acc[t][v] + hs[(size_t)grow * HID + col];  // + identity
        out[(size_t)grow * HID + col] = dinv[grow] * val;
      }
    }
  } else {
    // epilogue: gelu into LDS (lB[0] reused as gtile[32][256]), then LayerNorm
    float* gtile = &lB[0][0][0];
#pragma unroll
    for (int t = 0; t < 4; ++t) {
      const int col = col0 + t * 16 + l;
#pragma unroll
      for (int v = 0; v < 8; ++v) {
        const int rl = strip * 16 + v + rsel;
        const int grow = m0 + rl;
        float val = dinv[grow] * (acc[t][v] + hs[(size_t)grow * HID + col]);
        float g = 0.5f * val * (1.0f + erff(val * 0.70710678118654752f));
        gtile[rl * HID + col] = g;
      }
    }
    __syncthreads();
    const int r = tid >> 3, j = tid & 7;  // 8 threads per row
    float s = 0.f, s2 = 0.f;
#pragma unroll
    for (int c = 0; c < 32; ++c) {
      float gv = gtile[r * HID + (c << 3) + j];  // interleaved: distinct banks
      s += gv;
      s2 += gv * gv;
    }
    redS[r][j] = s;
    redS2[r][j] = s2;
    __syncthreads();
    if (tid < 32) {
      float S = 0.f, S2 = 0.f;
#pragma unroll
      for (int jj = 0; jj < 8; ++jj) {
        S += redS[tid][jj];
        S2 += redS2[tid][jj];
      }
      float mu = S * (1.0f / HID);
      float var = S2 * (1.0f / HID) - mu * mu;
      mu_s[tid] = mu;
      rstd_s[tid] = rsqrtf(var + 1e-6f);
    }
    __syncthreads();
    const float mu = mu_s[r], rstd = rstd_s[r];
#pragma unroll
    for (int c = 0; c < 32; ++c) {
      const int col = (c << 3) + j;
      float gv = gtile[r * HID + col];
      out[(size_t)(m0 + r) * HID + col] =
          (gv - mu) * rstd * gamma[col] + beta[col];
    }
  }
}

// ---------------------------------------------------------------------------
extern "C" void kernel_launch(void* const* d_in, const int* in_sizes, int n_in,
                              void* d_out, int out_size, void* d_ws,
                              size_t ws_size, hipStream_t stream) {
  (void)in_sizes; (void)n_in; (void)out_size; (void)ws_size;
  const float* x     = (const float*)d_in[0];
  const float* adj   = (const float*)d_in[1];
  const float* W1    = (const float*)d_in[2];
  const float* b1    = (const float*)d_in[3];
  const float* W2    = (const float*)d_in[4];
  const float* b2    = (const float*)d_in[5];
  const float* gamma = (const float*)d_in[6];
  const float* beta  = (const float*)d_in[7];
  float* out = (float*)d_out;

  char* ws = (char*)d_ws;
  float* dinv = (float*)ws;                                   // 32 KB
  float* hs1  = (float*)(ws + (32 << 10));                    // 8 MB
  float* ln   = hs1 + (size_t)NNODES * HID;                   // 8 MB
  float* hs2  = ln + (size_t)NNODES * HID;                    // 8 MB

  rowsum_dinv_kernel<<<NNODES, 256, 0, stream>>>(adj, dinv);
  dense_gemm_kernel<<<NNODES / 32, 256, 0, stream>>>(x, W1, b1, dinv, hs1, INDIM);
  adj_gemm_kernel<<<NNODES / 32, 256, 0, stream>>>(adj, hs1, dinv, gamma, beta, ln, 1);
  dense_gemm_kernel<<<NNODES / 32, 256, 0, stream>>>(ln, W2, b2, dinv, hs2, HID);
  adj_gemm_kernel<<<NNODES / 32, 256, 0, stream>>>(adj, hs2, dinv, gamma, beta, out, 0);
}